// ISONET_76175539962377
// MI455X (gfx1250) — compile-verified
//
#include <hip/hip_runtime.h>
#include <hip/hip_bf16.h>
#include <math.h>

typedef float v2f __attribute__((ext_vector_type(2)));
typedef float v8f __attribute__((ext_vector_type(8)));

#define N_PAIRS    256
#define N_GRAPHS   512
#define EDGES_PG   200
#define MAX_EDGES  256
#define N_NODES    32768
#define N_EDGES    102400
#define HD         128
#define EENC       64
#define MSGD       256
#define SKD        32
#define G3         384
#define PROP_STEPS 5
#define SK_ITERS   20
#define INV_TEMP   10.0f

// ---------------------------------------------------------------------------
// fp32 WMMA: D = A(16x4) x B(4x16) + C(16x16), exact fp32 math on matrix pipe
// A layout: lane l(0..15) holds row M=l; VGPR0/1 = K{0,1} (lanes 0-15) or
//           K{2,3} (lanes 16-31).  B mirrored with lane = N column.
// C/D: VGPR r -> M=r (lanes 0-15) / M=r+8 (lanes 16-31); N = lane&15.
// ---------------------------------------------------------------------------
__device__ __forceinline__ v8f wmma_f32(v2f a, v2f b, v8f c) {
  return __builtin_amdgcn_wmma_f32_16x16x4_f32(false, a, false, b, (short)0, c,
                                               false, false);
}

// Inner K-step shared by all GEMMs: one A b64 load feeds NT WMMAs.
template <int NT>
__device__ __forceinline__ void gemm_step(const float* __restrict__ apos,
                                          const float* __restrict__ wpos,
                                          int ldw, v8f (&acc)[NT]) {
  v2f a = *(const v2f*)apos;
#pragma unroll
  for (int t = 0; t < NT; ++t) {
    v2f b;
    b.x = wpos[(size_t)0 * ldw + t * 16];
    b.y = wpos[(size_t)1 * ldw + t * 16];
    acc[t] = wmma_f32(a, b, acc[t]);
  }
}

// ---------------------------------------------------------------------------
// Generic GEMM: Y[M,N] = act(X[M,K] @ W[K,N] + bias).
// One wave per 16 x (16*NT) tile.
// M % 16 == 0, K % 4 == 0, N % (16*NT) == 0 for all call sites.
// ---------------------------------------------------------------------------
template <int NT>
__global__ void ISONET_gemm_kernel(const float* __restrict__ X,
                                   const float* __restrict__ W,
                                   const float* __restrict__ bias,
                                   float* __restrict__ Y,
                                   int M, int K, int N, int relu) {
  const int lane = threadIdx.x;
  const int half = lane >> 4;
  const int l    = lane & 15;
  const int n0   = blockIdx.x * 16 * NT;
  const int m0   = blockIdx.y * 16;
  const float* __restrict__ arow = X + (size_t)(m0 + l) * K;
  const float* __restrict__ wp   = W + n0 + l;   // lane's B column base
  v8f acc[NT] = {};
  for (int kk = 0; kk < K; kk += 4) {
    const int kb = kk + half * 2;
    if ((kk & 15) == 0)  // prefetch next W K-panel (global_prefetch_b8)
      __builtin_prefetch(wp + (size_t)(kb + 16) * N, 0, 3);
    gemm_step<NT>(arow + kb, wp + (size_t)kb * N, N, acc);
  }
#pragma unroll
  for (int t = 0; t < NT; ++t) {
    const int col = n0 + t * 16 + l;
    const float bv = bias ? bias[col] : 0.0f;
#pragma unroll
    for (int r = 0; r < 8; ++r) {
      const int row = m0 + r + half * 8;
      float v = acc[t][r] + bv;
      if (relu) v = fmaxf(v, 0.0f);
      Y[(size_t)row * N + col] = v;
    }
  }
}

// ---------------------------------------------------------------------------
// Message MLP layer 1 with fused gather-concat: X = [h[src] | h[dst] | e]
// (swap=0 -> src=from,dst=to ; swap=1 -> src=to,dst=from). K=320, N=256, relu.
// 16x64 tile per wave. The K loop is split into the three concat segments so
// the hottest inner loop is branch-free straight-line loads + WMMA.
// ---------------------------------------------------------------------------
__global__ void ISONET_msg_l1_kernel(const float* __restrict__ h,
                                     const float* __restrict__ e,
                                     const int* __restrict__ from_idx,
                                     const int* __restrict__ to_idx,
                                     const float* __restrict__ W,
                                     const float* __restrict__ bias,
                                     float* __restrict__ hidden, int swap) {
  const int lane = threadIdx.x;
  const int half = lane >> 4;
  const int l    = lane & 15;
  const int n0   = blockIdx.x * 64;
  const int m0   = blockIdx.y * 16;
  const int m    = m0 + l;
  const int s_node = swap ? to_idx[m] : from_idx[m];
  const int d_node = swap ? from_idx[m] : to_idx[m];
  const float* __restrict__ srow = h + (size_t)s_node * HD;
  const float* __restrict__ drow = h + (size_t)d_node * HD;
  const float* __restrict__ erow = e + (size_t)m * EENC;
  const float* __restrict__ wp   = W + n0 + l;
  v8f acc[4] = {};
  // segment 1: k in [0,128)  -> h[src]
  for (int kk = 0; kk < HD; kk += 4) {
    const int kb = kk + half * 2;
    if ((kk & 15) == 0) __builtin_prefetch(wp + (size_t)(kb + 16) * MSGD, 0, 3);
    gemm_step<4>(srow + kb, wp + (size_t)kb * MSGD, MSGD, acc);
  }
  // segment 2: k in [128,256) -> h[dst]
  for (int kk = 0; kk < HD; kk += 4) {
    const int kb = kk + half * 2;
    if ((kk & 15) == 0)
      __builtin_prefetch(wp + (size_t)(HD + kb + 16) * MSGD, 0, 3);
    gemm_step<4>(drow + kb, wp + (size_t)(HD + kb) * MSGD, MSGD, acc);
  }
  // segment 3: k in [256,320) -> e
  for (int kk = 0; kk < EENC; kk += 4) {
    const int kb = kk + half * 2;
    if ((kk & 15) == 0)
      __builtin_prefetch(wp + (size_t)(2 * HD + kb + 16) * MSGD, 0, 3);
    gemm_step<4>(erow + kb, wp + (size_t)(2 * HD + kb) * MSGD, MSGD, acc);
  }
#pragma unroll
  for (int t = 0; t < 4; ++t) {
    const int col = n0 + t * 16 + l;
    const float bv = bias[col];
#pragma unroll
    for (int r = 0; r < 8; ++r) {
      const int row = m0 + r + half * 8;
      hidden[(size_t)row * MSGD + col] = fmaxf(acc[t][r] + bv, 0.0f);
    }
  }
}

// ---------------------------------------------------------------------------
// Message MLP layer 2 (K=N=256), 16x64 tile per wave. Epilogue fused:
//   mode 0: segment-sum  agg[idx[m]] += y  (global f32 atomics)
//   mode 1: out[m] = y        mode 2: out[m] += y   (e_final = fwd + rev)
// ---------------------------------------------------------------------------
__global__ void ISONET_msg_l2_kernel(const float* __restrict__ hidden,
                                     const float* __restrict__ W,
                                     const float* __restrict__ bias,
                                     const int* __restrict__ seg_idx,
                                     float* __restrict__ out, int mode) {
  const int lane = threadIdx.x;
  const int half = lane >> 4;
  const int l    = lane & 15;
  const int n0   = blockIdx.x * 64;
  const int m0   = blockIdx.y * 16;
  const float* __restrict__ arow = hidden + (size_t)(m0 + l) * MSGD;
  const float* __restrict__ wp   = W + n0 + l;
  v8f acc[4] = {};
  for (int kk = 0; kk < MSGD; kk += 4) {
    const int kb = kk + half * 2;
    if ((kk & 15) == 0) __builtin_prefetch(wp + (size_t)(kb + 16) * MSGD, 0, 3);
    gemm_step<4>(arow + kb, wp + (size_t)kb * MSGD, MSGD, acc);
  }
  float bv[4];
#pragma unroll
  for (int t = 0; t < 4; ++t) bv[t] = bias[n0 + t * 16 + l];
#pragma unroll
  for (int r = 0; r < 8; ++r) {
    const int row = m0 + r + half * 8;
    if (mode == 0) {
      const size_t base = (size_t)seg_idx[row] * MSGD;
#pragma unroll
      for (int t = 0; t < 4; ++t)
        atomicAdd(&out[base + n0 + t * 16 + l], acc[t][r] + bv[t]);
    } else if (mode == 1) {
#pragma unroll
      for (int t = 0; t < 4; ++t)
        out[(size_t)row * MSGD + n0 + t * 16 + l] = acc[t][r] + bv[t];
    } else {
#pragma unroll
      for (int t = 0; t < 4; ++t)
        out[(size_t)row * MSGD + n0 + t * 16 + l] += acc[t][r] + bv[t];
    }
  }
}

// ---------------------------------------------------------------------------
// GRU gate update (elementwise over [N_NODES, HD])
// ---------------------------------------------------------------------------
__global__ void ISONET_gru_kernel(const float* __restrict__ gi,
                                  const float* __restrict__ gh,
                                  float* __restrict__ h) {
  const int idx = blockIdx.x * blockDim.x + threadIdx.x;
  if (idx >= N_NODES * HD) return;
  const int node = idx >> 7;
  const int j    = idx & (HD - 1);
  const float* gin = gi + (size_t)node * G3;
  const float* ghn = gh + (size_t)node * G3;
  const float r = 1.0f / (1.0f + expf(-(gin[j] + ghn[j])));
  const float z = 1.0f / (1.0f + expf(-(gin[HD + j] + ghn[HD + j])));
  const float n = tanhf(gin[2 * HD + j] + r * ghn[2 * HD + j]);
  h[idx] = (1.0f - z) * n + z * h[idx];
}

// ---------------------------------------------------------------------------
// Build masked/padded mq, mc [N_PAIRS, MAX_EDGES, SKD] from t [N_EDGES, SKD]
// ---------------------------------------------------------------------------
__global__ void ISONET_build_mqmc_kernel(const float* __restrict__ t,
                                         float* __restrict__ mq,
                                         float* __restrict__ mc) {
  const int idx = blockIdx.x * blockDim.x + threadIdx.x;
  if (idx >= N_PAIRS * MAX_EDGES * SKD) return;
  const int f = idx & (SKD - 1);
  const int i = (idx >> 5) & (MAX_EDGES - 1);
  const int b = idx >> 13;
  float vq = 0.0f, vc = 0.0f;
  if (i < EDGES_PG) {
    vq = t[((size_t)(2 * b) * EDGES_PG + i) * SKD + f];
    vc = t[((size_t)(2 * b + 1) * EDGES_PG + i) * SKD + f];
  }
  mq[idx] = vq;
  mc[idx] = vc;
}

// ---------------------------------------------------------------------------
// la[b,i,j] = (mq[b,i,:] . mc[b,j,:]) / TEMP  (B transposed, K=32), 16x64 tile
// ---------------------------------------------------------------------------
__global__ void ISONET_la_gemm_kernel(const float* __restrict__ mq,
                                      const float* __restrict__ mc,
                                      float* __restrict__ la) {
  const int lane = threadIdx.x;
  const int half = lane >> 4;
  const int l    = lane & 15;
  const int n0   = blockIdx.x * 64;
  const int m0   = blockIdx.y * 16;
  const int b    = blockIdx.z;
  const float* __restrict__ A = mq + (size_t)b * MAX_EDGES * SKD;
  const float* __restrict__ B = mc + (size_t)b * MAX_EDGES * SKD;
  const float* __restrict__ arow = A + (size_t)(m0 + l) * SKD;
  v8f acc[4] = {};
  for (int kk = 0; kk < SKD; kk += 4) {
    const int kb = kk + half * 2;
    v2f a = *(const v2f*)(arow + kb);
#pragma unroll
    for (int t = 0; t < 4; ++t) {
      v2f bb = *(const v2f*)(B + (size_t)(n0 + t * 16 + l) * SKD + kb);  // B^T
      acc[t] = wmma_f32(a, bb, acc[t]);
    }
  }
  float* __restrict__ G = la + (size_t)b * MAX_EDGES * MAX_EDGES;
#pragma unroll
  for (int t = 0; t < 4; ++t) {
#pragma unroll
    for (int r = 0; r < 8; ++r) {
      const int row = m0 + r + half * 8;
      G[(size_t)row * MAX_EDGES + n0 + t * 16 + l] = acc[t][r] * INV_TEMP;
    }
  }
}

// ---------------------------------------------------------------------------
// Sinkhorn: whole 256x256 fp32 matrix lives in LDS (stride 257 -> 263 KB,
// fits CDNA5's 320 KB/WG LDS). 20 row+col log-sum-exp sweeps, then exp().
// One 256-thread workgroup (8 waves) per pair batch.
// ---------------------------------------------------------------------------
__global__ void ISONET_sinkhorn_kernel(float* __restrict__ la) {
  extern __shared__ float s[];
  const int b   = blockIdx.x;
  const int tid = threadIdx.x;      // 0..255
  const int LD  = MAX_EDGES + 1;    // bank-conflict padding
  float* __restrict__ g = la + (size_t)b * MAX_EDGES * MAX_EDGES;
  for (int idx = tid; idx < MAX_EDGES * MAX_EDGES; idx += 256) {
    const int i = idx >> 8, j = idx & 255;
    s[i * LD + j] = g[idx];
  }
  __syncthreads();
  for (int it = 0; it < SK_ITERS; ++it) {
    {  // axis=2 (row-wise over j): thread tid owns row tid
      float* row = s + tid * LD;
      float mx = -3.4e38f;
      for (int j = 0; j < MAX_EDGES; ++j) mx = fmaxf(mx, row[j]);
      float sum = 0.0f;
      for (int j = 0; j < MAX_EDGES; ++j) sum += expf(row[j] - mx);
      const float lse = mx + logf(sum);
      for (int j = 0; j < MAX_EDGES; ++j) row[j] -= lse;
    }
    __syncthreads();
    {  // axis=1 (col-wise over i): thread tid owns column tid
      float mx = -3.4e38f;
      for (int i = 0; i < MAX_EDGES; ++i) mx = fmaxf(mx, s[i * LD + tid]);
      float sum = 0.0f;
      for (int i = 0; i < MAX_EDGES; ++i) sum += expf(s[i * LD + tid] - mx);
      const float lse = mx + logf(sum);
      for (int i = 0; i < MAX_EDGES; ++i) s[i * LD + tid] -= lse;
    }
    __syncthreads();
  }
  for (int idx = tid; idx < MAX_EDGES * MAX_EDGES; idx += 256) {
    const int i = idx >> 8, j = idx & 255;
    g[idx] = expf(s[i * LD + j]);  // plan, in-place
  }
}

// ---------------------------------------------------------------------------
// r = plan[b] @ c[b] (K=256, c rows >=200 are zero), fused with
// out[b] -= sum relu(q - r). q/c read straight from e_final. 16x64 tile.
// ---------------------------------------------------------------------------
__global__ void ISONET_planc_kernel(const float* __restrict__ plan,
                                    const float* __restrict__ e_final,
                                    float* __restrict__ out) {
  const int lane = threadIdx.x;
  const int half = lane >> 4;
  const int l    = lane & 15;
  const int n0   = blockIdx.x * 64;   // feature tile
  const int m0   = blockIdx.y * 16;   // edge-row tile
  const int b    = blockIdx.z;
  const float* __restrict__ P  = plan + (size_t)b * MAX_EDGES * MAX_EDGES;
  const float* __restrict__ Cg = e_final + (size_t)(2 * b + 1) * EDGES_PG * MSGD;
  const float* __restrict__ Qg = e_final + (size_t)(2 * b) * EDGES_PG * MSGD;
  const float* __restrict__ arow = P + (size_t)(m0 + l) * MAX_EDGES;
  v8f acc[4] = {};
  for (int kk = 0; kk < MAX_EDGES; kk += 4) {
    const int kb = kk + half * 2;
    v2f a = *(const v2f*)(arow + kb);
#pragma unroll
    for (int t = 0; t < 4; ++t) {
      const int col = n0 + t * 16 + l;
      v2f bb;
      bb.x = (kb     < EDGES_PG) ? Cg[(size_t)kb * MSGD + col] : 0.0f;
      bb.y = (kb + 1 < EDGES_PG) ? Cg[(size_t)(kb + 1) * MSGD + col] : 0.0f;
      acc[t] = wmma_f32(a, bb, acc[t]);
    }
  }
  float part = 0.0f;
#pragma unroll
  for (int t = 0; t < 4; ++t) {
    const int col = n0 + t * 16 + l;
#pragma unroll
    for (int r = 0; r < 8; ++r) {
      const int row = m0 + r + half * 8;
      const float q = (row < EDGES_PG) ? Qg[(size_t)row * MSGD + col] : 0.0f;
      part += fmaxf(q - acc[t][r], 0.0f);
    }
  }
  atomicAdd(&out[b], -part);
}

__global__ void ISONET_zero_kernel(float* __restrict__ p, long n) {
  for (long i = blockIdx.x * (long)blockDim.x + threadIdx.x; i < n;
       i += (long)gridDim.x * blockDim.x)
    p[i] = 0.0f;
}

// ---------------------------------------------------------------------------
// Host driver
// ---------------------------------------------------------------------------
extern "C" void kernel_launch(void* const* d_in, const int* in_sizes, int n_in,
                              void* d_out, int out_size, void* d_ws,
                              size_t ws_size, hipStream_t stream) {
  (void)in_sizes; (void)n_in; (void)out_size; (void)ws_size;
  const float* node_features = (const float*)d_in[0];
  const float* edge_features = (const float*)d_in[1];
  const int*   from_idx      = (const int*)d_in[2];
  const int*   to_idx        = (const int*)d_in[3];
  const float* enc_node_w = (const float*)d_in[4];
  const float* enc_node_b = (const float*)d_in[5];
  const float* enc_edge_w = (const float*)d_in[6];
  const float* enc_edge_b = (const float*)d_in[7];
  const float* msg_w1  = (const float*)d_in[8];
  const float* msg_b1  = (const float*)d_in[9];
  const float* msg_w2  = (const float*)d_in[10];
  const float* msg_b2  = (const float*)d_in[11];
  const float* rmsg_w1 = (const float*)d_in[12];
  const float* rmsg_b1 = (const float*)d_in[13];
  const float* rmsg_w2 = (const float*)d_in[14];
  const float* rmsg_b2 = (const float*)d_in[15];
  const float* gru_w_ih = (const float*)d_in[16];
  const float* gru_b_ih = (const float*)d_in[17];
  const float* gru_w_hh = (const float*)d_in[18];
  const float* gru_b_hh = (const float*)d_in[19];
  const float* sk_w1 = (const float*)d_in[20];
  const float* sk_b1 = (const float*)d_in[21];
  const float* sk_w2 = (const float*)d_in[22];
  const float* sk_b2 = (const float*)d_in[23];
  float* out = (float*)d_out;

  // workspace carve-up (floats); post-loop buffers alias loop buffers
  float* ws = (float*)d_ws;
  float* h      = ws;                                      //  4,194,304
  float* e      = h      + (size_t)N_NODES * HD;           //  6,553,600
  float* hidden = e      + (size_t)N_EDGES * EENC;         // 26,214,400
  float* msgbuf = hidden + (size_t)N_EDGES * MSGD;         // 26,214,400
  float* agg    = msgbuf + (size_t)N_EDGES * MSGD;         //  8,388,608
  float* gi     = agg    + (size_t)N_NODES * MSGD;         // 12,582,912
  float* gh     = gi     + (size_t)N_NODES * G3;           // 12,582,912
  float* skh = agg;                                        // aliases (post-loop)
  float* t   = gi;
  float* mq  = gh;
  float* mc  = gh + (size_t)N_PAIRS * MAX_EDGES * SKD;
  float* la  = hidden;                                     // 16,777,216 fits

  const dim3 wv(32);

  // encoders
  ISONET_gemm_kernel<4><<<dim3(HD / 64, N_NODES / 16), wv, 0, stream>>>(
      node_features, enc_node_w, enc_node_b, h, N_NODES, 16, HD, 0);
  ISONET_gemm_kernel<4><<<dim3(EENC / 64, N_EDGES / 16), wv, 0, stream>>>(
      edge_features, enc_edge_w, enc_edge_b, e, N_EDGES, 16, EENC, 0);

  const dim3 msg_grid(MSGD / 64, N_EDGES / 16);
  for (int step = 0; step < PROP_STEPS; ++step) {
    ISONET_zero_kernel<<<2048, 256, 0, stream>>>(agg, (long)N_NODES * MSGD);
    // fwd messages -> segment-sum by to_idx
    ISONET_msg_l1_kernel<<<msg_grid, wv, 0, stream>>>(
        h, e, from_idx, to_idx, msg_w1, msg_b1, hidden, 0);
    ISONET_msg_l2_kernel<<<msg_grid, wv, 0, stream>>>(
        hidden, msg_w2, msg_b2, to_idx, agg, 0);
    // rev messages -> segment-sum by from_idx
    ISONET_msg_l1_kernel<<<msg_grid, wv, 0, stream>>>(
        h, e, from_idx, to_idx, rmsg_w1, rmsg_b1, hidden, 1);
    ISONET_msg_l2_kernel<<<msg_grid, wv, 0, stream>>>(
        hidden, rmsg_w2, rmsg_b2, from_idx, agg, 0);
    // GRU
    ISONET_gemm_kernel<4><<<dim3(G3 / 64, N_NODES / 16), wv, 0, stream>>>(
        agg, gru_w_ih, gru_b_ih, gi, N_NODES, MSGD, G3, 0);
    ISONET_gemm_kernel<4><<<dim3(G3 / 64, N_NODES / 16), wv, 0, stream>>>(
        h, gru_w_hh, gru_b_hh, gh, N_NODES, HD, G3, 0);
    ISONET_gru_kernel<<<(N_NODES * HD + 255) / 256, 256, 0, stream>>>(gi, gh, h);
  }

  // final messages: e_final = fwd + rev
  ISONET_msg_l1_kernel<<<msg_grid, wv, 0, stream>>>(
      h, e, from_idx, to_idx, msg_w1, msg_b1, hidden, 0);
  ISONET_msg_l2_kernel<<<msg_grid, wv, 0, stream>>>(
      hidden, msg_w2, msg_b2, nullptr, msgbuf, 1);
  ISONET_msg_l1_kernel<<<msg_grid, wv, 0, stream>>>(
      h, e, from_idx, to_idx, rmsg_w1, rmsg_b1, hidden, 1);
  ISONET_msg_l2_kernel<<<msg_grid, wv, 0, stream>>>(
      hidden, rmsg_w2, rmsg_b2, nullptr, msgbuf, 2);

  // SK transform t = relu(e_final @ w1 + b1) @ w2 + b2   (N=32 -> 16x32 tiles)
  ISONET_gemm_kernel<2><<<dim3(SKD / 32, N_EDGES / 16), wv, 0, stream>>>(
      msgbuf, sk_w1, sk_b1, skh, N_EDGES, MSGD, SKD, 1);
  ISONET_gemm_kernel<2><<<dim3(SKD / 32, N_EDGES / 16), wv, 0, stream>>>(
      skh, sk_w2, sk_b2, t, N_EDGES, SKD, SKD, 0);
  ISONET_build_mqmc_kernel<<<(N_PAIRS * MAX_EDGES * SKD + 255) / 256, 256, 0,
                             stream>>>(t, mq, mc);

  // la = mq @ mc^T / TEMP
  ISONET_la_gemm_kernel<<<dim3(MAX_EDGES / 64, MAX_EDGES / 16, N_PAIRS), wv, 0,
                          stream>>>(mq, mc, la);

  // Sinkhorn entirely in 263 KB of LDS per workgroup (CDNA5: 320 KB/WG)
  const int sk_lds = (MAX_EDGES + 1) * MAX_EDGES * (int)sizeof(float);
  (void)hipFuncSetAttribute((const void*)ISONET_sinkhorn_kernel,
                            hipFuncAttributeMaxDynamicSharedMemorySize, sk_lds);
  ISONET_sinkhorn_kernel<<<N_PAIRS, 256, sk_lds, stream>>>(la);

  // out[b] = -sum relu(q - plan @ c), fused GEMM + reduction
  ISONET_zero_kernel<<<1, 256, 0, stream>>>(out, N_PAIRS);
  ISONET_planc_kernel<<<dim3(MSGD / 64, MAX_EDGES / 16, N_PAIRS), wv, 0,
                        stream>>>(la, msgbuf, out);
}